// JpegCompression_50070728737401
// MI455X (gfx1250) — compile-verified
//
#include <hip/hip_runtime.h>

typedef float v2f __attribute__((ext_vector_type(2)));
typedef float v8f __attribute__((ext_vector_type(8)));

#define STR  68            // padded LDS row stride (floats): 272B = 17*16 -> aligned + conflict-free
#define LDSC (16 * STR)    // per-channel LDS floats (16 block-columns)

// ---------------------------------------------------------------------------
// Setup: build the combined per-channel 64x64 operator
//   T_c[(pr,pc),(qr,qc)] = sum_{i,j} Di[pr,i]*D[i,qr] * mask_c[i,j] * Di[pc,j]*D[j,qc]
// and store it pre-swizzled in the WMMA A-operand layout:
//   ws[c][t][k][lane][0..1] = T_c[16t + (lane&15)][4k + 2*(lane>>4) + j]
// ---------------------------------------------------------------------------
__global__ void JpegBuildT_kernel(const float* __restrict__ Dm,
                                  const float* __restrict__ Di,
                                  const float* __restrict__ mask,
                                  float* __restrict__ ws)
{
    int idx = blockIdx.x * blockDim.x + threadIdx.x;
    if (idx >= 3 * 64 * 64) return;
    int c = idx >> 12;
    int m = (idx >> 6) & 63;        // output element (pr,pc)
    int q = idx & 63;               // input element (qr,qc)
    int pr = m >> 3, pc = m & 7, qr = q >> 3, qc = q & 7;
    const float* mk = mask + (long)c * 512 * 512;   // mask is tiled: top-left 8x8 block
    float acc = 0.0f;
    for (int i = 0; i < 8; ++i) {
        float ai = Di[pr * 8 + i] * Dm[i * 8 + qr];
        float inner = 0.0f;
        for (int j = 0; j < 8; ++j)
            inner += mk[i * 512 + j] * Di[pc * 8 + j] * Dm[j * 8 + qc];
        acc += ai * inner;
    }
    int t = m >> 4, ml = m & 15;
    int k = q >> 2, h = (q >> 1) & 1, jj = q & 1;
    ws[c * 4096 + t * 1024 + k * 64 + (h * 16 + ml) * 2 + jj] = acc;
}

// ---------------------------------------------------------------------------
// Main kernel: one WG = 8-row x 128-col strip = 16 blocks x 3 channels.
// 12 waves: wave w -> channel w>>2, output-row tile w&3.
// ---------------------------------------------------------------------------
__global__ __launch_bounds__(384)
void JpegCompression_50070728737401_kernel(const float* __restrict__ img,
                                           const float* __restrict__ T,
                                           float* __restrict__ out)
{
    __shared__ __align__(16) float sIn[3 * LDSC];
    __shared__ __align__(16) float sOut[3 * LDSC];

    const int g   = blockIdx.x;
    const int b   = g >> 8;               // batch
    const int s   = g & 255;              // strip within image
    const int r0  = (s >> 2) << 3;        // strip row start
    const int c0  = (s & 3) << 7;         // strip col start
    const int tid = threadIdx.x;
    const long plane = 512l * 512l;

    // Stage 1: RGB -> YUV, store to LDS in B-operand layout sIn[ch][n*STR + rr]
    for (int p = tid; p < 1024; p += 384) {
        int row = p >> 7, col = p & 127;
        long base = ((long)(b * 3) * 512 + (r0 + row)) * 512 + (c0 + col);
        float r  = img[base];
        float gc = img[base + plane];
        float bc = img[base + 2 * plane];
        float y =  0.299f   * r + 0.587f   * gc + 0.114f   * bc;
        float u = -0.14713f * r - 0.28886f * gc + 0.436f   * bc;
        float v =  0.615f   * r - 0.51499f * gc - 0.10001f * bc;
        int n  = col >> 3;                      // block column within strip
        int rr = (row << 3) | (col & 7);        // raster index within 8x8 block
        sIn[0 * LDSC + n * STR + rr] = y;
        sIn[1 * LDSC + n * STR + rr] = u;
        sIn[2 * LDSC + n * STR + rr] = v;
    }
    __syncthreads();

    // Stage 2: Out[16t..16t+15][0..15] = T_c[rows] * X   (K = 64, 16 WMMAs)
    {
        int w    = tid >> 5;        // wave 0..11
        int lane = tid & 31;
        int ch   = w >> 2;          // YUV channel
        int t    = w & 3;           // 16-row output tile
        int h    = lane >> 4;
        int nl   = lane & 15;

        const float* aBase = T + ((ch * 4 + t) * 16) * 64;      // + k*64 + lane*2
        const float* sB    = sIn + ch * LDSC + nl * STR;        // + 4k + 2h

        v8f acc = {};
        #pragma unroll
        for (int k = 0; k < 16; ++k) {
            v2f a  = *(const v2f*)(aBase + k * 64 + lane * 2);  // A: 16x4 fp32 chunk
            v2f bb = *(const v2f*)(sB + 4 * k + 2 * h);         // B: 4x16 fp32 chunk
            acc = __builtin_amdgcn_wmma_f32_16x16x4_f32(
                /*neg_a=*/false, a, /*neg_b=*/false, bb,
                /*c_mod=*/(short)0, acc, /*reuse_a=*/false, /*reuse_b=*/false);
        }
        // D layout: reg j, lane half h -> output element p = 16t + 8h + j, column nl
        float* oB = sOut + ch * LDSC + nl * STR + t * 16 + h * 8;
        *(float4*)(oB)     = make_float4(acc[0], acc[1], acc[2], acc[3]);
        *(float4*)(oB + 4) = make_float4(acc[4], acc[5], acc[6], acc[7]);
    }
    __syncthreads();

    // Stage 3: YUV -> RGB, coalesced store
    for (int p = tid; p < 1024; p += 384) {
        int row = p >> 7, col = p & 127;
        int n  = col >> 3;
        int rr = (row << 3) | (col & 7);
        float y = sOut[0 * LDSC + n * STR + rr];
        float u = sOut[1 * LDSC + n * STR + rr];
        float v = sOut[2 * LDSC + n * STR + rr];
        float r  = y + 1.13983f * v;
        float gc = y - 0.39465f * u - 0.5806f  * v;
        float bc = y + 2.03211f * u;
        long base = ((long)(b * 3) * 512 + (r0 + row)) * 512 + (c0 + col);
        out[base]             = r;
        out[base + plane]     = gc;
        out[base + 2 * plane] = bc;
    }
}

// ---------------------------------------------------------------------------
extern "C" void kernel_launch(void* const* d_in, const int* in_sizes, int n_in,
                              void* d_out, int out_size, void* d_ws, size_t ws_size,
                              hipStream_t stream)
{
    (void)in_sizes; (void)n_in; (void)out_size; (void)ws_size;
    const float* img  = (const float*)d_in[0];   // (32,3,512,512) fp32
    const float* Dd   = (const float*)d_in[1];   // (8,8) DCT
    const float* Didc = (const float*)d_in[2];   // (8,8) IDCT
    const float* mask = (const float*)d_in[3];   // (3,512,512) tiled masks
    float* T = (float*)d_ws;                     // 3*64*64 fp32 = 48 KB

    JpegBuildT_kernel<<<48, 256, 0, stream>>>(Dd, Didc, mask, T);
    JpegCompression_50070728737401_kernel<<<8192, 384, 0, stream>>>(img, T, (float*)d_out);
}